// SelectiveScan_89558658056576
// MI455X (gfx1250) — compile-verified
//
#include <hip/hip_runtime.h>
#include <hip/hip_bf16.h>
#include <stdint.h>

// Problem constants (from the reference)
#define B_  2
#define L_  2048
#define D_  1024
#define N_  16
#define NT  256          // threads per workgroup (8 waves of 32)
#define T_  (L_ / NT)    // 8 consecutive l-steps per thread

// Selective scan, 3-phase chunked formulation with LDS-cached decay factors.
// Exact fp32 simplification of the reference:
//   g[l,n] = exp(exp(dt*A[d,n]))   (g = 1 at l==0, from the shift/pad)
//   S[l,n] = g*S[l-1,n] + (dt*u)*B[l,n]
//   y[l,d] = sum_n S[l,n]*C[l,n] + u*D[d]
// Phase 1 computes g ONCE (2 v_exp_f32 per element, the dominant cost),
// caches it in LDS; phase 3 replays the recurrence from the corrected
// incoming state using the cached g -> 128M total transcendentals, not 256M.
__global__ __launch_bounds__(NT) void selscan_kernel(
    const float* __restrict__ u,  const float* __restrict__ dt,
    const float* __restrict__ A,  const float* __restrict__ Bm,
    const float* __restrict__ Cm, const float* __restrict__ Dv,
    float* __restrict__ out)
{
    __shared__ float g_s[L_ * N_]; // 128 KB: cached decay factors g[l][n]
    __shared__ float u_s[L_];      //   8 KB: staged via async-to-LDS gather
    __shared__ float dt_s[L_];     //   8 KB: staged via async-to-LDS gather
    __shared__ float Ga[NT][N_];   //  16 KB: per-chunk decay product
    __shared__ float Bc[NT][N_];   //  16 KB: per-chunk local scan -> exclusive prefix

    const int b  = blockIdx.x / D_;
    const int d  = blockIdx.x % D_;
    const int t  = threadIdx.x;
    const int l0 = t * T_;

    // ---- Phase 0: async gather of the 4KB-strided u/dt columns into LDS ----
    {
        const unsigned col_byte =
            (unsigned)(((unsigned)(b * L_) * D_ + (unsigned)d) * sizeof(float));
        #pragma unroll
        for (int i = 0; i < T_; ++i) {
            const int l = l0 + i;
            const unsigned goff   = col_byte + (unsigned)(l * D_ * (int)sizeof(float));
            const unsigned lds_u  = (unsigned)(uintptr_t)&u_s[l];
            const unsigned lds_dt = (unsigned)(uintptr_t)&dt_s[l];
            asm volatile("global_load_async_to_lds_b32 %0, %1, %2 offset:0"
                         :: "v"(lds_u),  "v"(goff), "s"(u)  : "memory");
            asm volatile("global_load_async_to_lds_b32 %0, %1, %2 offset:0"
                         :: "v"(lds_dt), "v"(goff), "s"(dt) : "memory");
        }
        asm volatile("s_wait_asynccnt 0" ::: "memory");
    }
    __syncthreads();

    // Per-channel constants: A[d, 0:16] and D[d]
    float Arow[N_];
    {
        const float4* A4 = reinterpret_cast<const float4*>(A + (size_t)d * N_);
        #pragma unroll
        for (int q = 0; q < 4; ++q) {
            float4 v = A4[q];
            Arow[4*q+0] = v.x; Arow[4*q+1] = v.y; Arow[4*q+2] = v.z; Arow[4*q+3] = v.w;
        }
    }
    const float Dd = Dv[d];
    const float* Bbase = Bm + (size_t)(b * L_) * N_;
    const float* Cbase = Cm + (size_t)(b * L_) * N_;

    // ---- Phase 1: per-chunk affine summary; compute & cache g in LDS ----
    float S[N_], G[N_];
    float ureg[T_], dtureg[T_];          // private: only owner thread reuses them
    #pragma unroll
    for (int n = 0; n < N_; ++n) { S[n] = 0.0f; G[n] = 1.0f; }

    for (int i = 0; i < T_; ++i) {
        const int l = l0 + i;
        const float dtv = dt_s[l];
        const float uv  = u_s[l];
        const float dtu = dtv * uv;
        ureg[i]   = uv;
        dtureg[i] = dtu;

        float Bv[N_];
        const float4* B4 = reinterpret_cast<const float4*>(Bbase + (size_t)l * N_);
        #pragma unroll
        for (int q = 0; q < 4; ++q) {
            float4 v = B4[q];
            Bv[4*q+0] = v.x; Bv[4*q+1] = v.y; Bv[4*q+2] = v.z; Bv[4*q+3] = v.w;
        }

        float gv[N_];
        #pragma unroll
        for (int n = 0; n < N_; ++n) {
            float g = __expf(__expf(dtv * Arow[n]));
            if (l == 0) g = 1.0f;                 // shifted[0] == 0
            gv[n] = g;
            S[n] = fmaf(g, S[n], dtu * Bv[n]);
            G[n] *= g;
        }
        // cache the decay row for phase 3 (ds_store_b128 x4)
        float4* grow = reinterpret_cast<float4*>(g_s + (size_t)l * N_);
        #pragma unroll
        for (int q = 0; q < 4; ++q) {
            float4 v; v.x = gv[4*q+0]; v.y = gv[4*q+1]; v.z = gv[4*q+2]; v.w = gv[4*q+3];
            grow[q] = v;
        }
    }
    #pragma unroll
    for (int n = 0; n < N_; ++n) { Ga[t][n] = G[n]; Bc[t][n] = S[n]; }
    __syncthreads();

    // ---- Phase 2: compose chunk summaries; leave exclusive prefix state in Bc ----
    if (t < N_) {
        const int n = t;
        float s = 0.0f;
        for (int c = 0; c < NT; ++c) {
            const float a  = Ga[c][n];
            const float bb = Bc[c][n];
            Bc[c][n] = s;              // state entering chunk c
            s = fmaf(a, s, bb);        // state after chunk c
        }
    }
    __syncthreads();

    // ---- Phase 3: replay chunk from corrected state using cached g ----
    #pragma unroll
    for (int n = 0; n < N_; ++n) S[n] = Bc[t][n];

    float* outb = out + (size_t)(b * L_) * D_ + d;
    for (int i = 0; i < T_; ++i) {
        const int l = l0 + i;
        const float dtu = dtureg[i];

        float Bv[N_], Cv[N_], gv[N_];
        const float4* B4 = reinterpret_cast<const float4*>(Bbase + (size_t)l * N_);
        const float4* C4 = reinterpret_cast<const float4*>(Cbase + (size_t)l * N_);
        const float4* G4 = reinterpret_cast<const float4*>(g_s + (size_t)l * N_);
        #pragma unroll
        for (int q = 0; q < 4; ++q) {
            float4 vb = B4[q], vc = C4[q], vg = G4[q];
            Bv[4*q+0] = vb.x; Bv[4*q+1] = vb.y; Bv[4*q+2] = vb.z; Bv[4*q+3] = vb.w;
            Cv[4*q+0] = vc.x; Cv[4*q+1] = vc.y; Cv[4*q+2] = vc.z; Cv[4*q+3] = vc.w;
            gv[4*q+0] = vg.x; gv[4*q+1] = vg.y; gv[4*q+2] = vg.z; gv[4*q+3] = vg.w;
        }

        float y = ureg[i] * Dd;
        #pragma unroll
        for (int n = 0; n < N_; ++n) {
            S[n] = fmaf(gv[n], S[n], dtu * Bv[n]);
            y = fmaf(S[n], Cv[n], y);
        }
        outb[(size_t)l * D_] = y;
    }
}

extern "C" void kernel_launch(void* const* d_in, const int* in_sizes, int n_in,
                              void* d_out, int out_size, void* d_ws, size_t ws_size,
                              hipStream_t stream) {
    (void)in_sizes; (void)n_in; (void)out_size; (void)d_ws; (void)ws_size;
    const float* u  = (const float*)d_in[0];
    const float* dt = (const float*)d_in[1];
    const float* A  = (const float*)d_in[2];
    const float* Bm = (const float*)d_in[3];
    const float* Cm = (const float*)d_in[4];
    const float* Dv = (const float*)d_in[5];
    float* out = (float*)d_out;

    dim3 grid(B_ * D_);   // one workgroup per (b, d) channel
    dim3 block(NT);
    hipLaunchKernelGGL(selscan_kernel, grid, block, 0, stream,
                       u, dt, A, Bm, Cm, Dv, out);
}